// TypilusModel_64561948393778
// MI455X (gfx1250) — compile-verified
//
#include <hip/hip_runtime.h>
#include <hip/hip_bf16.h>

typedef float v2f __attribute__((ext_vector_type(2)));
typedef float v8f __attribute__((ext_vector_type(8)));

#define NNODES   100000
#define F_IN     128
#define HID      64
#define FC_DIM   128
#define NCLASSES 100
#define NGRAPHS  256

// ---------------- utility: fill ----------------
__global__ void fill_kernel(float* __restrict__ p, float v, int n) {
  int i = blockIdx.x * blockDim.x + threadIdx.x;
  if (i < n) p[i] = v;
}

// ---------------- degree (with self-loops pre-seeded to 1.0) ----------------
__global__ void deg_kernel(const int* __restrict__ dst, float* __restrict__ deg, int nE) {
  int e = blockIdx.x * blockDim.x + threadIdx.x;
  if (e < nE) atomicAdd(&deg[dst[e]], 1.0f);
}

__global__ void rsqrt_kernel(float* __restrict__ d, int n) {
  int i = blockIdx.x * blockDim.x + threadIdx.x;
  if (i < n) d[i] = rsqrtf(d[i]);
}

// ---------------- weight repack: Wp[k/2][n][2] = { W[2m][n], W[2m+1][n] } ----------------
// Lets the WMMA B-fragment (needs rows kk and kk+1 at the same column) be one b64 load.
__global__ void repack_W(const float* __restrict__ W, float* __restrict__ Wp, int K, int N) {
  int i = blockIdx.x * blockDim.x + threadIdx.x;
  if (i >= K * N) return;
  int k = i / N;
  int n = i - k * N;
  Wp[(size_t)(k >> 1) * N * 2 + n * 2 + (k & 1)] = W[i];
}

// ---------------- fp32 WMMA GEMM: C[M x N] = A[M x K] * W[K x N] ----------------
// One wave per 16x16 output tile, K stepped by 4 with V_WMMA_F32_16X16X4_F32.
// K and N are compile-time so all A/B loads and C stores fold to base+imm offsets.
// A frag (16x4):  lanes 0-15 hold rows M=0..15 with K={0,1}; lanes 16-31 K={2,3}.
//   -> each lane loads a float4 (K..K+3) of its row and selects its half.
// B frag (4x16):  VGPR0 = K rows {0|2}, VGPR1 = {1|3}; one v2f load from repacked Wp.
// C/D (16x16 f32): VGPR r -> M=r (lanes 0-15), M=r+8 (lanes 16-31), N=lane&15.
template <int K, int N>
__global__ void gemm_wmma_f32(const float* __restrict__ A, const float* __restrict__ Wp,
                              float* __restrict__ C, int M) {
  int wave   = (blockIdx.x * blockDim.x + threadIdx.x) >> 5;
  int lane   = threadIdx.x & 31;
  constexpr int tilesN = N >> 4;
  int tm = wave / tilesN;
  int tn = wave % tilesN;
  if (tm * 16 >= M) return;            // wave-uniform guard (EXEC all-ones for WMMA)

  int half = lane >> 4;                // 0: K pair {k,k+1}; 1: K pair {k+2,k+3}
  int l    = lane & 15;

  v8f acc = {};
  const float4* Arow = (const float4*)(A + (size_t)(tm * 16 + l) * K);
  const v2f*    Bp   = (const v2f*)Wp + (size_t)half * N + tn * 16 + l;
#pragma unroll
  for (int k4 = 0; k4 < K / 4; ++k4) {
    float4 av = Arow[k4];              // global_load_b128, imm offset
    v2f a;
    a.x = half ? av.z : av.x;
    a.y = half ? av.w : av.y;
    v2f b = Bp[(size_t)(2 * k4) * N];  // global_load_b64, imm offset (N constexpr)
    acc = __builtin_amdgcn_wmma_f32_16x16x4_f32(
        /*neg_a=*/false, a, /*neg_b=*/false, b,
        /*c_mod=*/(short)0, acc, /*reuse_a=*/false, /*reuse_b=*/false);
  }

  float* Crow = C + (size_t)(tm * 16 + half * 8) * N + tn * 16 + l;
#pragma unroll
  for (int r = 0; r < 8; ++r) Crow[(size_t)r * N] = acc[r];  // imm-offset stores
}

// ---------------- edge scatter: agg[dst] += xw[src] * dinv[src]*dinv[dst] ----------------
// thread = edge*16 + chunk; each thread moves one float4 (16B) of the 64-float row.
__global__ void scatter_kernel(const float* __restrict__ xw,
                               const int* __restrict__ src, const int* __restrict__ dst,
                               const float* __restrict__ dinv,
                               float* __restrict__ agg, int nE) {
  long long tid = (long long)blockIdx.x * blockDim.x + threadIdx.x;
  if (tid >= (long long)nE * 16) return;
  int e = (int)(tid >> 4);
  int c = (int)(tid & 15);
  int s = src[e];
  int d = dst[e];
  float w = dinv[s] * dinv[d];
  const float4 v = *(const float4*)(xw + (size_t)s * HID + c * 4);
  float* o = agg + (size_t)d * HID + c * 4;
  atomicAdd(o + 0, v.x * w);
  atomicAdd(o + 1, v.y * w);
  atomicAdd(o + 2, v.z * w);
  atomicAdd(o + 3, v.w * w);
}

// ---------------- fused self-loop + bias + relu (in place on agg) ----------------
__global__ void bias_relu_self(float* __restrict__ agg, const float* __restrict__ xw,
                               const float* __restrict__ dinv, const float* __restrict__ b,
                               int n /* = NNODES*HID */) {
  int i = blockIdx.x * blockDim.x + threadIdx.x;
  if (i >= n) return;
  int node = i >> 6;          // HID == 64
  int f    = i & 63;
  float di = dinv[node];
  float v  = agg[i] + xw[i] * di * di + b[f];
  agg[i]   = fmaxf(v, 0.0f);
}

// ---------------- pooling: atomic sums per graph ----------------
__global__ void pool_kernel(const float* __restrict__ h, const int* __restrict__ batch,
                            float* __restrict__ sums, float* __restrict__ counts, int n) {
  int i = blockIdx.x * blockDim.x + threadIdx.x;
  if (i >= n) return;
  int node = i >> 6;
  int f    = i & 63;
  int g    = batch[node];
  atomicAdd(&sums[(size_t)g * HID + f], h[i]);
  if (f == 0) atomicAdd(&counts[g], 1.0f);
}

// ---------------- FC head: one 128-thread block per graph ----------------
__global__ void fc_kernel(const float* __restrict__ sums, const float* __restrict__ counts,
                          const float* __restrict__ fcW1, const float* __restrict__ fcb1,
                          const float* __restrict__ fcW2, const float* __restrict__ fcb2,
                          float* __restrict__ out) {
  __shared__ float pooled[HID];
  __shared__ float t[FC_DIM];
  int g = blockIdx.x;
  int j = threadIdx.x;                 // 0..127
  if (j < HID) {
    float c = counts[g];
    pooled[j] = sums[(size_t)g * HID + j] / fmaxf(c, 1.0f);
  }
  __syncthreads();
  float acc = fcb1[j];
#pragma unroll 8
  for (int k = 0; k < HID; ++k) acc += pooled[k] * fcW1[(size_t)k * FC_DIM + j];
  t[j] = fmaxf(acc, 0.0f);
  __syncthreads();
  if (j < NCLASSES) {
    float acc2 = fcb2[j];
#pragma unroll 8
    for (int k = 0; k < FC_DIM; ++k) acc2 += t[k] * fcW2[(size_t)k * NCLASSES + j];
    out[(size_t)g * NCLASSES + j] = acc2;
  }
}

static inline size_t align256(size_t x) { return (x + 255) & ~(size_t)255; }

extern "C" void kernel_launch(void* const* d_in, const int* in_sizes, int n_in,
                              void* d_out, int out_size, void* d_ws, size_t ws_size,
                              hipStream_t stream) {
  const float* x    = (const float*)d_in[0];   // [N, 128]
  const float* W1   = (const float*)d_in[1];   // [128, 64]
  const float* b1   = (const float*)d_in[2];   // [64]
  const float* W2   = (const float*)d_in[3];   // [64, 64]
  const float* b2   = (const float*)d_in[4];   // [64]
  const float* fcW1 = (const float*)d_in[5];   // [64, 128]
  const float* fcb1 = (const float*)d_in[6];   // [128]
  const float* fcW2 = (const float*)d_in[7];   // [128, 100]
  const float* fcb2 = (const float*)d_in[8];   // [100]
  const int*   ei   = (const int*)d_in[9];     // [2, E]
  const int*   batch= (const int*)d_in[10];    // [N]
  float* out = (float*)d_out;                  // [256, 100]

  const int N = in_sizes[0] / F_IN;            // 100000
  const int E = in_sizes[9] / 2;               // 3200000
  const int* src = ei;
  const int* dst = ei + E;

  // workspace layout
  char* ws = (char*)d_ws;
  size_t off = 0;
  float* bufX   = (float*)(ws + off); off = align256(off + (size_t)N * HID * 4);   // xw
  float* bufY   = (float*)(ws + off); off = align256(off + (size_t)N * HID * 4);   // agg / h
  float* dinv   = (float*)(ws + off); off = align256(off + (size_t)N * 4);
  float* sums   = (float*)(ws + off); off = align256(off + (size_t)NGRAPHS * HID * 4);
  float* counts = (float*)(ws + off); off = align256(off + (size_t)NGRAPHS * 4);
  float* Wp1    = (float*)(ws + off); off = align256(off + (size_t)F_IN * HID * 4);
  float* Wp2    = (float*)(ws + off); off = align256(off + (size_t)HID * HID * 4);
  (void)ws_size; (void)n_in; (void)out_size;

  const int T = 256;
  const int nNH = N * HID;
  const int gNH = (nNH + T - 1) / T;
  const int gN  = (N + T - 1) / T;
  const int gE  = (E + T - 1) / T;
  long long scatterThreads = (long long)E * 16;
  const int gScat = (int)((scatterThreads + T - 1) / T);
  const int wavesGemm = (N / 16) * (HID / 16);           // 25000 waves
  const int gGemm = (wavesGemm * 32 + T - 1) / T;        // 8 waves/block

  // 0. repack weights for b64 B-fragment loads
  repack_W<<<(F_IN * HID + T - 1) / T, T, 0, stream>>>(W1, Wp1, F_IN, HID);
  repack_W<<<(HID * HID + T - 1) / T, T, 0, stream>>>(W2, Wp2, HID, HID);

  // 1. degree with self-loops -> dinv
  fill_kernel<<<gN, T, 0, stream>>>(dinv, 1.0f, N);
  deg_kernel<<<gE, T, 0, stream>>>(dst, dinv, E);
  rsqrt_kernel<<<gN, T, 0, stream>>>(dinv, N);

  // 2. layer 1: xw = x @ W1  (K=128) — fp32 WMMA
  gemm_wmma_f32<F_IN, HID><<<gGemm, T, 0, stream>>>(x, Wp1, bufX, N);
  fill_kernel<<<gNH, T, 0, stream>>>(bufY, 0.0f, nNH);
  scatter_kernel<<<gScat, T, 0, stream>>>(bufX, src, dst, dinv, bufY, E);
  bias_relu_self<<<gNH, T, 0, stream>>>(bufY, bufX, dinv, b1, nNH);

  // 3. layer 2: xw = h1 @ W2 (K=64) — fp32 WMMA
  gemm_wmma_f32<HID, HID><<<gGemm, T, 0, stream>>>(bufY, Wp2, bufX, N);
  fill_kernel<<<gNH, T, 0, stream>>>(bufY, 0.0f, nNH);
  scatter_kernel<<<gScat, T, 0, stream>>>(bufX, src, dst, dinv, bufY, E);
  bias_relu_self<<<gNH, T, 0, stream>>>(bufY, bufX, dinv, b2, nNH);

  // 4. mean pool + MLP head
  fill_kernel<<<(NGRAPHS * HID + T - 1) / T, T, 0, stream>>>(sums, 0.0f, NGRAPHS * HID);
  fill_kernel<<<1, T, 0, stream>>>(counts, 0.0f, NGRAPHS);
  pool_kernel<<<gNH, T, 0, stream>>>(bufY, batch, sums, counts, nNH);
  fc_kernel<<<NGRAPHS, FC_DIM, 0, stream>>>(sums, counts, fcW1, fcb1, fcW2, fcb2, out);
}